// HistogramModel_60868276519083
// MI455X (gfx1250) — compile-verified
//
#include <hip/hip_runtime.h>
#include <hip/hip_bf16.h>

typedef __attribute__((ext_vector_type(2))) float v2f;
typedef __attribute__((ext_vector_type(8))) float v8f;
typedef __attribute__((ext_vector_type(4))) int   i4v;

// ---------------- Problem constants ----------------
#define BATCH   64
#define CHAN    3
#define HWSZ    50176          // 224*224
#define KDIM    150528         // 3*224*224
#define EMB     512
#define HIN_DIM 812            // 512 + 300
#define HID     400
#define BINS    100

// base GEMM tiling
#define BK      32
#define LDSS    (BK + 4)       // 36: keeps b128 LDS rows 16B-aligned, bank-conflict-free frags
#define NTILE_W 128
#define KSPLIT  56
#define KCHUNK  (KDIM / KSPLIT)   // 2688, multiple of BK (84 tiles)

// CDNA5 async global->LDS copy path (guarded: falls back to reg-staged copy)
#if defined(__has_builtin)
#if __has_builtin(__builtin_amdgcn_global_load_async_to_lds_b128)
#define HAVE_ASYNC_LDS 1
#endif
#endif
#ifndef HAVE_ASYNC_LDS
#define HAVE_ASYNC_LDS 0
#endif

#if HAVE_ASYNC_LDS
// param1: addrspace(1) int4*  (prints as "int __vector_size__(16) * __device__")
// param2: addrspace(3) int4*
#define AS_G(p) ((__attribute__((address_space(1))) i4v*)(const void*)(p))
#define AS_L(p) ((__attribute__((address_space(3))) i4v*)(void*)(p))
#endif

// ---------------- 1) per-plane min/max ----------------
__global__ __launch_bounds__(256) void minmax_kernel(const float* __restrict__ x,
                                                     float* __restrict__ mn,
                                                     float* __restrict__ mx) {
    __shared__ float smin[256], smax[256];
    const int p = blockIdx.x;                    // 0..191 (B*C planes)
    const float* base = x + (size_t)p * HWSZ;
    float lo = 3.402823466e38f, hi = -3.402823466e38f;
    for (int i = threadIdx.x; i < HWSZ; i += 256) {
        float v = base[i];
        lo = fminf(lo, v);
        hi = fmaxf(hi, v);
    }
    smin[threadIdx.x] = lo; smax[threadIdx.x] = hi;
    __syncthreads();
    for (int s = 128; s > 0; s >>= 1) {
        if (threadIdx.x < s) {
            smin[threadIdx.x] = fminf(smin[threadIdx.x], smin[threadIdx.x + s]);
            smax[threadIdx.x] = fmaxf(smax[threadIdx.x], smax[threadIdx.x + s]);
        }
        __syncthreads();
    }
    if (threadIdx.x == 0) { mn[p] = smin[0]; mx[p] = smax[0]; }
}

// ---------------- 2) per-plane 100-bin histogram -> hin[:,512+...] ----------------
__global__ __launch_bounds__(256) void hist_kernel(const float* __restrict__ x,
                                                   const float* __restrict__ mn,
                                                   const float* __restrict__ mx,
                                                   float* __restrict__ hin) {
    __shared__ unsigned int bins[BINS];
    const int p = blockIdx.x;                    // plane = b*3 + c
    const int b = p / CHAN, c = p % CHAN;
    if (threadIdx.x < BINS) bins[threadIdx.x] = 0u;
    __syncthreads();
    const float lo = mn[p], hi = mx[p];
    const float width = (hi > lo) ? (hi - lo) : 1.0f;
    const float inv_scaled = (float)BINS / width;
    const float* base = x + (size_t)p * HWSZ;
    for (int i = threadIdx.x; i < HWSZ; i += 256) {
        float v = base[i];
        int idx = (int)floorf((v - lo) * inv_scaled);
        idx = idx < 0 ? 0 : (idx > BINS - 1 ? BINS - 1 : idx);
        atomicAdd(&bins[idx], 1u);               // LDS ds_add_u32 (deterministic value)
    }
    __syncthreads();
    if (threadIdx.x < BINS) {
        hin[(size_t)b * HIN_DIM + EMB + c * BINS + threadIdx.x] = (float)bins[threadIdx.x];
    }
}

// ---------------- 3) base GEMM: partial[ks] = x[64,Kc] @ W[128,Kc]^T ----------------
// grid = (4 n-blocks of 128, KSPLIT), block = 256 (8 waves). Double-buffered LDS;
// async copy (global_load_async_to_lds_b128 + s_wait_asynccnt) when available.
__global__ __launch_bounds__(256) void base_gemm_kernel(const float* __restrict__ x,
                                                        const float* __restrict__ W,
                                                        float* __restrict__ partial) {
    __shared__ __align__(16) float Xs[2][BATCH * LDSS];
    __shared__ __align__(16) float Ws[2][NTILE_W * LDSS];
    const int nblk = blockIdx.x;                 // 0..3
    const int ks   = blockIdx.y;                 // 0..KSPLIT-1
    const int t    = threadIdx.x;
    const int wave = t >> 5;
    const int lane = t & 31;
    const int mrow = lane & 15;
    const int kh   = lane >> 4;                  // half-wave K offset
    const int k0   = ks * KCHUNK;
    const int NT   = KCHUNK / BK;                // 84

    // chunk coords: X has 512 float4 chunks (2/thread), W has 1024 (4/thread)
    const int xr0 = t >> 3,           xc0 = (t & 7) << 2;
    const int xr1 = (t + 256) >> 3,   xc1 = ((t + 256) & 7) << 2;
    const int wr0 = t >> 3,           wc0 = (t & 7) << 2;
    const int wr1 = (t + 256) >> 3,   wc1 = ((t + 256) & 7) << 2;
    const int wr2 = (t + 512) >> 3,   wc2 = ((t + 512) & 7) << 2;
    const int wr3 = (t + 768) >> 3,   wc3 = ((t + 768) & 7) << 2;
    const float* Wb0 = W + (size_t)(nblk * NTILE_W) * KDIM;

    v8f acc0 = {}, acc1 = {}, acc2 = {}, acc3 = {};

#define COMPUTE_TILE(cur)                                                                   \
    do {                                                                                    \
        const float* Xb = &Xs[cur][0];                                                      \
        const float* Wq = &Ws[cur][0];                                                      \
        _Pragma("unroll")                                                                   \
        for (int kk = 0; kk < BK; kk += 4) {                                                \
            const int ko = kk + 2 * kh;                                                     \
            v2f bf = *(const v2f*)&Wq[(wave * 16 + mrow) * LDSS + ko];                      \
            v2f a0 = *(const v2f*)&Xb[( 0 + mrow) * LDSS + ko];                             \
            v2f a1 = *(const v2f*)&Xb[(16 + mrow) * LDSS + ko];                             \
            v2f a2 = *(const v2f*)&Xb[(32 + mrow) * LDSS + ko];                             \
            v2f a3 = *(const v2f*)&Xb[(48 + mrow) * LDSS + ko];                             \
            acc0 = __builtin_amdgcn_wmma_f32_16x16x4_f32(false, a0, false, bf, (short)0, acc0, false, false); \
            acc1 = __builtin_amdgcn_wmma_f32_16x16x4_f32(false, a1, false, bf, (short)0, acc1, false, false); \
            acc2 = __builtin_amdgcn_wmma_f32_16x16x4_f32(false, a2, false, bf, (short)0, acc2, false, false); \
            acc3 = __builtin_amdgcn_wmma_f32_16x16x4_f32(false, a3, false, bf, (short)0, acc3, false, false); \
        }                                                                                   \
    } while (0)

#if HAVE_ASYNC_LDS
    // ---- async path: global_load_async_to_lds_b128 + s_wait_asynccnt, 2x LDS buffers ----
#define STAGE_ASYNC(bufi, kbase)                                                                             \
    do {                                                                                                     \
        __builtin_amdgcn_global_load_async_to_lds_b128(AS_G(x   + (size_t)xr0 * KDIM + (kbase) + xc0),       \
                                                       AS_L(&Xs[bufi][xr0 * LDSS + xc0]), 0, 0);             \
        __builtin_amdgcn_global_load_async_to_lds_b128(AS_G(x   + (size_t)xr1 * KDIM + (kbase) + xc1),       \
                                                       AS_L(&Xs[bufi][xr1 * LDSS + xc1]), 0, 0);             \
        __builtin_amdgcn_global_load_async_to_lds_b128(AS_G(Wb0 + (size_t)wr0 * KDIM + (kbase) + wc0),       \
                                                       AS_L(&Ws[bufi][wr0 * LDSS + wc0]), 0, 0);             \
        __builtin_amdgcn_global_load_async_to_lds_b128(AS_G(Wb0 + (size_t)wr1 * KDIM + (kbase) + wc1),       \
                                                       AS_L(&Ws[bufi][wr1 * LDSS + wc1]), 0, 0);             \
        __builtin_amdgcn_global_load_async_to_lds_b128(AS_G(Wb0 + (size_t)wr2 * KDIM + (kbase) + wc2),       \
                                                       AS_L(&Ws[bufi][wr2 * LDSS + wc2]), 0, 0);             \
        __builtin_amdgcn_global_load_async_to_lds_b128(AS_G(Wb0 + (size_t)wr3 * KDIM + (kbase) + wc3),       \
                                                       AS_L(&Ws[bufi][wr3 * LDSS + wc3]), 0, 0);             \
    } while (0)

    STAGE_ASYNC(0, k0);
    for (int tile = 0; tile < NT; ++tile) {
        const int cur = tile & 1;
        if (tile + 1 < NT) {
            STAGE_ASYNC(cur ^ 1, k0 + (tile + 1) * BK);
            asm volatile("s_wait_asynccnt 0x6" ::: "memory");  // current tile's 6 have landed
        } else {
            asm volatile("s_wait_asynccnt 0x0" ::: "memory");
        }
        __syncthreads();
        COMPUTE_TILE(cur);
        __syncthreads();
    }
#else
    // ---- fallback: register-double-buffered global loads + ds_store_b128 ----
    float4 rx0, rx1, rw0, rw1, rw2, rw3;
#define LOAD_REGS(kbase)                                                        \
    do {                                                                        \
        rx0 = *(const float4*)(x   + (size_t)xr0 * KDIM + (kbase) + xc0);       \
        rx1 = *(const float4*)(x   + (size_t)xr1 * KDIM + (kbase) + xc1);       \
        rw0 = *(const float4*)(Wb0 + (size_t)wr0 * KDIM + (kbase) + wc0);       \
        rw1 = *(const float4*)(Wb0 + (size_t)wr1 * KDIM + (kbase) + wc1);       \
        rw2 = *(const float4*)(Wb0 + (size_t)wr2 * KDIM + (kbase) + wc2);       \
        rw3 = *(const float4*)(Wb0 + (size_t)wr3 * KDIM + (kbase) + wc3);       \
    } while (0)
#define STORE_REGS(bufi)                                                        \
    do {                                                                        \
        *(float4*)&Xs[bufi][xr0 * LDSS + xc0] = rx0;                            \
        *(float4*)&Xs[bufi][xr1 * LDSS + xc1] = rx1;                            \
        *(float4*)&Ws[bufi][wr0 * LDSS + wc0] = rw0;                            \
        *(float4*)&Ws[bufi][wr1 * LDSS + wc1] = rw1;                            \
        *(float4*)&Ws[bufi][wr2 * LDSS + wc2] = rw2;                            \
        *(float4*)&Ws[bufi][wr3 * LDSS + wc3] = rw3;                            \
    } while (0)

    LOAD_REGS(k0);
    for (int tile = 0; tile < NT; ++tile) {
        const int cur = tile & 1;
        STORE_REGS(cur);
        if (tile + 1 < NT) LOAD_REGS(k0 + (tile + 1) * BK);   // in flight during compute
        __syncthreads();
        COMPUTE_TILE(cur);
        __syncthreads();
    }
#endif

    float* pout = partial + (size_t)ks * BATCH * EMB;
    const int col = nblk * NTILE_W + wave * 16 + mrow;
#pragma unroll
    for (int g = 0; g < 8; ++g) {
        const int r = g + 8 * kh;
        pout[(size_t)( 0 + r) * EMB + col] = acc0[g];
        pout[(size_t)(16 + r) * EMB + col] = acc1[g];
        pout[(size_t)(32 + r) * EMB + col] = acc2[g];
        pout[(size_t)(48 + r) * EMB + col] = acc3[g];
    }
}

// ---------------- 4) split-K reduce (+bias) -> hin[:,0:512] ----------------
__global__ __launch_bounds__(256) void reduce_kernel(const float* __restrict__ partial,
                                                     const float* __restrict__ bias,
                                                     float* __restrict__ hin) {
    const int i = blockIdx.x * 256 + threadIdx.x;    // < 64*512
    float s = 0.f;
    for (int p = 0; p < KSPLIT; ++p) s += partial[(size_t)p * BATCH * EMB + i];
    const int b = i >> 9, n = i & 511;
    hin[(size_t)b * HIN_DIM + n] = s + bias[n];
}

// ---------------- 5) generic WMMA GEMM: out = relu(A[64,K] @ W[N,K]^T + bias) ----------------
// grid = (ceil(N/16), 4), block = 32 (one wave). K multiple of 4 (812/400/600 all are).
__global__ __launch_bounds__(32) void gemm_relu_kernel(const float* __restrict__ A,
                                                       const float* __restrict__ W,
                                                       const float* __restrict__ bias,
                                                       float* __restrict__ out,
                                                       int N, int K) {
    const int n0 = blockIdx.x * 16, m0 = blockIdx.y * 16;
    const int lane = threadIdx.x;
    const int mrow = lane & 15;
    const int kh   = lane >> 4;
    const int ncol = n0 + mrow;
    const bool nok = (ncol < N);

    const float* ap = A + (size_t)(m0 + mrow) * K + 2 * kh;
    const float* wp = W + (size_t)(nok ? ncol : 0) * K + 2 * kh;

    v8f acc = {};
    for (int k = 0; k < K; k += 4) {
        v2f a = *(const v2f*)(ap + k);
        v2f b = *(const v2f*)(wp + k);
        if (!nok) { b.x = 0.f; b.y = 0.f; }
        acc = __builtin_amdgcn_wmma_f32_16x16x4_f32(false, a, false, b, (short)0, acc, false, false);
    }
    if (nok) {
        const float bv = bias[ncol];
#pragma unroll
        for (int g = 0; g < 8; ++g) {
            const int row = m0 + g + 8 * kh;
            float v = acc[g] + bv;
            out[(size_t)row * N + ncol] = v > 0.f ? v : 0.f;
        }
    }
}

// ---------------- 6) in-place row softmax on the three heads ----------------
__device__ inline void softmax_row(float* p, int n) {
    float m = p[0];
    for (int i = 1; i < n; ++i) m = fmaxf(m, p[i]);
    float s = 0.f;
    for (int i = 0; i < n; ++i) { float e = __expf(p[i] - m); p[i] = e; s += e; }
    float inv = 1.0f / s;
    for (int i = 0; i < n; ++i) p[i] *= inv;
}

__global__ __launch_bounds__(64) void softmax_kernel(float* __restrict__ out) {
    const int b = threadIdx.x;
    if (b >= BATCH) return;
    softmax_row(out + (size_t)b * 10, 10);               // x_st
    softmax_row(out + 640 + (size_t)b * 4, 4);           // x_age
    softmax_row(out + 896 + (size_t)b * 2, 2);           // x_gender
}

// ---------------- launch ----------------
extern "C" void kernel_launch(void* const* d_in, const int* in_sizes, int n_in,
                              void* d_out, int out_size, void* d_ws, size_t ws_size,
                              hipStream_t stream) {
    const float* x      = (const float*)d_in[0];
    const float* base_W = (const float*)d_in[1];
    const float* base_b = (const float*)d_in[2];
    const float* hW     = (const float*)d_in[3];
    const float* hb     = (const float*)d_in[4];
    const float* stW1   = (const float*)d_in[5];
    const float* stb1   = (const float*)d_in[6];
    const float* stW2   = (const float*)d_in[7];
    const float* stb2   = (const float*)d_in[8];
    const float* ageW1  = (const float*)d_in[9];
    const float* ageb1  = (const float*)d_in[10];
    const float* ageW2  = (const float*)d_in[11];
    const float* ageb2  = (const float*)d_in[12];
    const float* genW1  = (const float*)d_in[13];
    const float* genb1  = (const float*)d_in[14];
    const float* genW2  = (const float*)d_in[15];
    const float* genb2  = (const float*)d_in[16];
    float* out = (float*)d_out;

    float* ws = (float*)d_ws;
    size_t off = 0;
    float* partial = ws + off; off += (size_t)KSPLIT * BATCH * EMB;   // 1,835,008
    float* hin     = ws + off; off += (size_t)BATCH * HIN_DIM;        // 51,968
    float* hbuf    = ws + off; off += (size_t)BATCH * HID;            // 25,600
    float* t_st    = ws + off; off += (size_t)BATCH * 600;
    float* t_age   = ws + off; off += (size_t)BATCH * 400;
    float* t_gen   = ws + off; off += (size_t)BATCH * 400;
    float* mn      = ws + off; off += BATCH * CHAN;
    float* mx      = ws + off; off += BATCH * CHAN;

    // histogram branch
    minmax_kernel<<<BATCH * CHAN, 256, 0, stream>>>(x, mn, mx);
    hist_kernel<<<BATCH * CHAN, 256, 0, stream>>>(x, mn, mx, hin);

    // base embedding GEMM (split-K WMMA, async double-buffered) + deterministic reduce
    base_gemm_kernel<<<dim3(4, KSPLIT), 256, 0, stream>>>(x, base_W, partial);
    reduce_kernel<<<(BATCH * EMB) / 256, 256, 0, stream>>>(partial, base_b, hin);

    // shared hidden layer: h = relu(hin @ hW^T + hb)   [64,400], K=812
    gemm_relu_kernel<<<dim3(HID / 16, 4), 32, 0, stream>>>(hin, hW, hb, hbuf, HID, HIN_DIM);

    // heads: relu GEMM -> relu GEMM (logits) -> softmax (in place on d_out)
    gemm_relu_kernel<<<dim3(38, 4), 32, 0, stream>>>(hbuf, stW1, stb1, t_st, 600, HID);
    gemm_relu_kernel<<<dim3(1, 4),  32, 0, stream>>>(t_st, stW2, stb2, out, 10, 600);

    gemm_relu_kernel<<<dim3(25, 4), 32, 0, stream>>>(hbuf, ageW1, ageb1, t_age, 400, HID);
    gemm_relu_kernel<<<dim3(1, 4),  32, 0, stream>>>(t_age, ageW2, ageb2, out + 640, 4, 400);

    gemm_relu_kernel<<<dim3(25, 4), 32, 0, stream>>>(hbuf, genW1, genb1, t_gen, 400, HID);
    gemm_relu_kernel<<<dim3(1, 4),  32, 0, stream>>>(t_gen, genW2, genb2, out + 896, 2, 400);

    softmax_kernel<<<1, 64, 0, stream>>>(out);
    (void)in_sizes; (void)n_in; (void)out_size; (void)ws_size;
}